// Dual_GraphSAGE_60610578481668
// MI455X (gfx1250) — compile-verified
//
#include <hip/hip_runtime.h>
#include <math.h>

#define NN 50000
#define NE 800000
#define C  128
#define AH 64

typedef float v2f __attribute__((ext_vector_type(2)));
typedef float v4f __attribute__((ext_vector_type(4)));
typedef float v8f __attribute__((ext_vector_type(8)));

__device__ __forceinline__ float wave_sum32(float v) {
#pragma unroll
    for (int off = 16; off > 0; off >>= 1) v += __shfl_xor(v, off, 32);
    return v;
}

// ---------------------------------------------------------------------------
// Zero a float buffer (n must be a multiple of 4).
__global__ void zero_f4(float* __restrict__ p, int n4) {
    int i = blockIdx.x * blockDim.x + threadIdx.x;
    if (i < n4) {
        v4f z = {0.f, 0.f, 0.f, 0.f};
        *(v4f*)(p + (size_t)i * 4) = z;
    }
}

// ---------------------------------------------------------------------------
// Per-edge cosine / euclidean similarity. One wave32 per edge; lane owns a
// float4 slice of the 128-wide feature row. Optionally accumulates degree.
__global__ void edge_sim(const float* __restrict__ x,
                         const int* __restrict__ row,
                         const int* __restrict__ col,
                         float* __restrict__ out_cos,   // may be null
                         float* __restrict__ out_eud,   // may be null
                         float* __restrict__ deg) {     // may be null
    const int lane = threadIdx.x & 31;
    const int e = blockIdx.x * (blockDim.x >> 5) + (threadIdx.x >> 5);
    if (e >= NE) return;
    const int r = row[e];
    const int c = col[e];
    v4f a = *(const v4f*)(x + (size_t)r * C + lane * 4);
    v4f b = *(const v4f*)(x + (size_t)c * C + lane * 4);
    float dot = a.x * b.x + a.y * b.y + a.z * b.z + a.w * b.w;
    float na  = a.x * a.x + a.y * a.y + a.z * a.z + a.w * a.w;
    float nb  = b.x * b.x + b.y * b.y + b.z * b.z + b.w * b.w;
    v4f d = a - b;
    d.x += 1e-6f; d.y += 1e-6f; d.z += 1e-6f; d.w += 1e-6f;
    float dd  = d.x * d.x + d.y * d.y + d.z * d.z + d.w * d.w;
    dot = wave_sum32(dot);
    na  = wave_sum32(na);
    nb  = wave_sum32(nb);
    dd  = wave_sum32(dd);
    if (lane == 0) {
        if (out_cos) out_cos[e] = dot / fmaxf(sqrtf(na) * sqrtf(nb), 1e-8f);
        if (out_eud) out_eud[e] = sqrtf(dd);
        if (deg) atomicAdd(deg + r, 1.0f);
    }
}

// deg -> 1/max(deg,1)
__global__ void invdeg_k(float* __restrict__ deg, int n) {
    int i = blockIdx.x * blockDim.x + threadIdx.x;
    if (i < n) deg[i] = 1.0f / fmaxf(deg[i], 1.0f);
}

// ---------------------------------------------------------------------------
// agg[row[e]] += ew[e] * x[col[e]]  — one wave per edge, f32 global atomics
__global__ void scatter_agg(const float* __restrict__ x,
                            const int* __restrict__ row,
                            const int* __restrict__ col,
                            const float* __restrict__ ew,
                            float* __restrict__ agg) {
    const int lane = threadIdx.x & 31;
    const int e = blockIdx.x * (blockDim.x >> 5) + (threadIdx.x >> 5);
    if (e >= NE) return;
    const float w = ew[e];
    const int r = row[e];
    const int c = col[e];
    v4f v = *(const v4f*)(x + (size_t)c * C + lane * 4);
    float* dst = agg + (size_t)r * C + lane * 4;
    atomicAdd(dst + 0, v.x * w);
    atomicAdd(dst + 1, v.y * w);
    atomicAdd(dst + 2, v.z * w);
    atomicAdd(dst + 3, v.w * w);
}

// ---------------------------------------------------------------------------
// out = [relu]( (agg*invdeg) @ Wl + bl + xin @ Wr )
// One 16x16 f32 WMMA tile per wave; 8 waves/block span the 128 output cols;
// gridDim.x * 16 spans the rows exactly (50000 = 3125*16) -> no divergence,
// EXEC is all-ones for every v_wmma as the ISA requires.
__global__ void sage_gemm(const float* __restrict__ agg,
                          const float* __restrict__ invdeg,
                          const float* __restrict__ xin,
                          const float* __restrict__ Wl,
                          const float* __restrict__ bl,
                          const float* __restrict__ Wr,
                          float* __restrict__ out,
                          int relu) {
    const int lane = threadIdx.x & 31;
    const int wave = threadIdx.x >> 5;          // 0..7 -> column tile
    const int m0   = blockIdx.x * 16;
    const int n0   = wave * 16;
    const int half = lane >> 4;                 // 0: K%4 in {0,1}  1: {2,3}
    const int l15  = lane & 15;
    const int arow = m0 + l15;                  // A-frag row (M = lane%16)
    const int koff = half * 2;
    const float id = invdeg[arow];
    const float* aggr = agg + (size_t)arow * C;
    const float* xr   = xin + (size_t)arow * C;
    v8f c = {0.f, 0.f, 0.f, 0.f, 0.f, 0.f, 0.f, 0.f};
#pragma unroll 4
    for (int k0 = 0; k0 < C; k0 += 4) {
        const int k = k0 + koff;
        // A-frag (16x4 f32): v0=A[M,k], v1=A[M,k+1]  -> one b64 load
        v2f a1 = *(const v2f*)(aggr + k);
        a1.x *= id; a1.y *= id;
        // B-frag (4x16 f32): lane=N, v0=B[k,N], v1=B[k+1,N]
        v2f b1;
        b1.x = Wl[(size_t)(k + 0) * C + n0 + l15];
        b1.y = Wl[(size_t)(k + 1) * C + n0 + l15];
        c = __builtin_amdgcn_wmma_f32_16x16x4_f32(false, a1, false, b1,
                                                  (short)0, c, false, false);
        v2f a2 = *(const v2f*)(xr + k);
        v2f b2;
        b2.x = Wr[(size_t)(k + 0) * C + n0 + l15];
        b2.y = Wr[(size_t)(k + 1) * C + n0 + l15];
        c = __builtin_amdgcn_wmma_f32_16x16x4_f32(false, a2, false, b2,
                                                  (short)0, c, false, false);
    }
    const float bias = bl[n0 + l15];
    // C/D layout: VGPR j -> M = m0 + j + half*8, N = n0 + (lane&15)
#pragma unroll
    for (int j = 0; j < 8; ++j) {
        const int m = m0 + j + half * 8;
        float v = c[j] + bias;
        if (relu) v = fmaxf(v, 0.f);
        out[(size_t)m * C + n0 + l15] = v;
    }
}

// ---------------------------------------------------------------------------
// Attention pooling over the 4 branch embeddings.
// attW1 (128x64, 32KB) staged in LDS; one wave per node, 8 nodes per wave.
__global__ void attention(const float* __restrict__ x1,
                          const float* __restrict__ x2,
                          const float* __restrict__ x3,
                          const float* __restrict__ x4,
                          const float* __restrict__ attW1,
                          const float* __restrict__ attb1,
                          const float* __restrict__ attW2,
                          float* __restrict__ emb) {
    __shared__ float Ws[C * AH];
    __shared__ float b1s[AH];
    __shared__ float w2s[AH];
    for (int i = threadIdx.x; i < C * AH; i += blockDim.x) Ws[i] = attW1[i];
    if (threadIdx.x < AH) {
        b1s[threadIdx.x] = attb1[threadIdx.x];
        w2s[threadIdx.x] = attW2[threadIdx.x];
    }
    __syncthreads();

    const int lane = threadIdx.x & 31;
    const int wave = threadIdx.x >> 5;
    const float* zs[4] = {x1, x2, x3, x4};

    for (int t = 0; t < 8; ++t) {
        const int node = blockIdx.x * 64 + wave * 8 + t;
        if (node >= NN) continue;
        v4f zreg[4];
        float wsc[4];
#pragma unroll
        for (int zi = 0; zi < 4; ++zi) {
            v4f z = *(const v4f*)(zs[zi] + (size_t)node * C + lane * 4);
            zreg[zi] = z;
            // lane computes hidden units `lane` and `lane+32`
            float h0 = b1s[lane];
            float h1 = b1s[lane + 32];
#pragma unroll 4
            for (int kq = 0; kq < 32; ++kq) {
                const float z0 = __shfl(z.x, kq, 32);
                const float z1 = __shfl(z.y, kq, 32);
                const float z2 = __shfl(z.z, kq, 32);
                const float z3 = __shfl(z.w, kq, 32);
                const int k = kq * 4;
                h0 += z0 * Ws[(k + 0) * AH + lane] + z1 * Ws[(k + 1) * AH + lane] +
                      z2 * Ws[(k + 2) * AH + lane] + z3 * Ws[(k + 3) * AH + lane];
                h1 += z0 * Ws[(k + 0) * AH + lane + 32] + z1 * Ws[(k + 1) * AH + lane + 32] +
                      z2 * Ws[(k + 2) * AH + lane + 32] + z3 * Ws[(k + 3) * AH + lane + 32];
            }
            h0 = tanhf(h0);
            h1 = tanhf(h1);
            wsc[zi] = wave_sum32(h0 * w2s[lane] + h1 * w2s[lane + 32]);
        }
        // softmax over the 4 branches (replicated on every lane)
        float mx = fmaxf(fmaxf(wsc[0], wsc[1]), fmaxf(wsc[2], wsc[3]));
        float bsum = 0.f, beta[4];
#pragma unroll
        for (int zi = 0; zi < 4; ++zi) { beta[zi] = __expf(wsc[zi] - mx); bsum += beta[zi]; }
        const float inv = 1.0f / bsum;
        v4f acc = {0.f, 0.f, 0.f, 0.f};
#pragma unroll
        for (int zi = 0; zi < 4; ++zi) {
            const float bz = beta[zi] * inv;
            acc.x += bz * zreg[zi].x; acc.y += bz * zreg[zi].y;
            acc.z += bz * zreg[zi].z; acc.w += bz * zreg[zi].w;
        }
        *(v4f*)(emb + (size_t)node * C + lane * 4) = acc;
    }
}

// ---------------------------------------------------------------------------
extern "C" void kernel_launch(void* const* d_in, const int* in_sizes, int n_in,
                              void* d_out, int out_size, void* d_ws, size_t ws_size,
                              hipStream_t stream) {
    const float* x     = (const float*)d_in[0];
    const int*   row   = (const int*)d_in[1];
    const int*   col   = (const int*)d_in[2];
    const float* Wl0   = (const float*)d_in[3];
    const float* bl0   = (const float*)d_in[4];
    const float* Wr0   = (const float*)d_in[5];
    const float* Wl1   = (const float*)d_in[6];
    const float* bl1   = (const float*)d_in[7];
    const float* Wr1   = (const float*)d_in[8];
    const float* attW1 = (const float*)d_in[9];
    const float* attb1 = (const float*)d_in[10];
    const float* attW2 = (const float*)d_in[11];

    // Outputs: (emb, x3, x4) concatenated
    float* emb = (float*)d_out;
    float* x3  = emb + (size_t)NN * C;
    float* x4  = x3  + (size_t)NN * C;

    // Workspace layout
    float* ws    = (float*)d_ws;
    float* e_cos = ws;                       ws += NE;
    float* e_eud = ws;                       ws += NE;
    float* e2    = ws;                       ws += NE;
    float* deg   = ws;                       ws += NN;
    float* agg   = ws;                       ws += (size_t)NN * C;
    float* x1    = ws;                       ws += (size_t)NN * C;
    float* x2    = ws;                       ws += (size_t)NN * C;

    const int edgeBlocks = NE / 8;           // 1 wave per edge, 8 waves/block
    const int gemmBlocks = NN / 16;          // 16 rows per block (exact)
    const int zeroAggBlk = (NN * C / 4 + 255) / 256;
    const int zeroDegBlk = (NN / 4 + 255) / 256;

    // degree (shared by all 4 SAGE calls) + first-layer edge weights
    zero_f4<<<zeroDegBlk, 256, 0, stream>>>(deg, NN / 4);
    edge_sim<<<edgeBlocks, 256, 0, stream>>>(x, row, col, e_cos, e_eud, deg);
    invdeg_k<<<(NN + 255) / 256, 256, 0, stream>>>(deg, NN);

    // ---- cosine branch, layer 0 -> x3 (relu)
    zero_f4<<<zeroAggBlk, 256, 0, stream>>>(agg, NN * C / 4);
    scatter_agg<<<edgeBlocks, 256, 0, stream>>>(x, row, col, e_cos, agg);
    sage_gemm<<<gemmBlocks, 256, 0, stream>>>(agg, deg, x, Wl0, bl0, Wr0, x3, 1);

    // cosine on x3 -> layer 1 -> x1
    edge_sim<<<edgeBlocks, 256, 0, stream>>>(x3, row, col, e2, nullptr, nullptr);
    zero_f4<<<zeroAggBlk, 256, 0, stream>>>(agg, NN * C / 4);
    scatter_agg<<<edgeBlocks, 256, 0, stream>>>(x3, row, col, e2, agg);
    sage_gemm<<<gemmBlocks, 256, 0, stream>>>(agg, deg, x3, Wl1, bl1, Wr1, x1, 0);

    // ---- euclidean branch, layer 0 -> x4 (relu)
    zero_f4<<<zeroAggBlk, 256, 0, stream>>>(agg, NN * C / 4);
    scatter_agg<<<edgeBlocks, 256, 0, stream>>>(x, row, col, e_eud, agg);
    sage_gemm<<<gemmBlocks, 256, 0, stream>>>(agg, deg, x, Wl0, bl0, Wr0, x4, 1);

    // euclidean on x4 -> layer 1 -> x2
    edge_sim<<<edgeBlocks, 256, 0, stream>>>(x4, row, col, nullptr, e2, nullptr);
    zero_f4<<<zeroAggBlk, 256, 0, stream>>>(agg, NN * C / 4);
    scatter_agg<<<edgeBlocks, 256, 0, stream>>>(x4, row, col, e2, agg);
    sage_gemm<<<gemmBlocks, 256, 0, stream>>>(agg, deg, x4, Wl1, bl1, Wr1, x2, 0);

    // ---- attention pooling over (x1, x2, x3, x4) -> emb
    attention<<<(NN + 63) / 64, 256, 0, stream>>>(x1, x2, x3, x4,
                                                  attW1, attb1, attW2, emb);
}